// GCN_6605659701859
// MI455X (gfx1250) — compile-verified
//
#include <hip/hip_runtime.h>
#include <hip/hip_bf16.h>

typedef float v2f __attribute__((ext_vector_type(2)));
typedef float v8f __attribute__((ext_vector_type(8)));

static constexpr int kNodes = 100000;
static constexpr int kEdges = 3200000;
static constexpr int kTiles = (kNodes + 15) / 16;   // 6250 exact

// ---------------------------------------------------------------- degrees
__global__ void gcn_init_deg(float* __restrict__ deg_out,
                             float* __restrict__ deg_in) {
    int i = blockIdx.x * blockDim.x + threadIdx.x;
    if (i < kNodes) {               // self-loop contributes 1 to both degrees
        deg_out[i] = 1.0f;
        deg_in[i]  = 1.0f;
    }
}

// int64 -> int32 edge compaction + degree accumulation (one streaming pass)
__global__ void gcn_prep_edges(const long long* __restrict__ e64,
                               int* __restrict__ src32,
                               int* __restrict__ dst32,
                               float* __restrict__ deg_out,
                               float* __restrict__ deg_in) {
    int e = blockIdx.x * blockDim.x + threadIdx.x;
    if (e < kEdges) {
        int s = (int)__builtin_nontemporal_load(&e64[e]);
        int d = (int)__builtin_nontemporal_load(&e64[kEdges + e]);
        src32[e] = s;
        dst32[e] = d;
        atomicAdd(&deg_out[s], 1.0f);
        atomicAdd(&deg_in[d], 1.0f);
    }
}

__global__ void gcn_norm(float* __restrict__ deg_out,
                         float* __restrict__ deg_in) {
    int i = blockIdx.x * blockDim.x + threadIdx.x;
    if (i < kNodes) {
        deg_out[i] = rsqrtf(fmaxf(deg_out[i], 1.0f));  // norm_src
        deg_in[i]  = rsqrtf(fmaxf(deg_in[i],  1.0f));  // norm_dst
    }
}

// ------------------------------------------------- layer 1: (X @ W1) * norm_src
// One wave per 16-node tile.  D = A(16x16 nodes x feat) * B(16x16) via
// four chained V_WMMA_F32_16X16X4_F32 (K split 4+4+4+4), full fp32.
__global__ void gcn_layer1_wmma(const float* __restrict__ X,
                                const float* __restrict__ W1,
                                const float* __restrict__ norm_src,
                                float* __restrict__ h1,
                                float* __restrict__ agg1) {
    const int wave = threadIdx.x >> 5;
    const int lane = threadIdx.x & 31;
    const int t    = blockIdx.x * (blockDim.x >> 5) + wave;
    if (t >= kTiles) return;                 // wave-uniform -> EXEC stays full
    const int hi = lane >> 4;                // half-wave selects K pair
    const int m  = lane & 15;                // A: row M / B,D: column N

    v8f c = {};
#pragma unroll
    for (int kk = 0; kk < 4; ++kk) {
        const int k0 = kk * 4 + hi * 2;      // K index of this lane's pair
        v2f av, bv;
        av[0] = X[(t * 16 + m) * 16 + k0 + 0];       // A 16x4: lane holds X[m,k0..k0+1]
        av[1] = X[(t * 16 + m) * 16 + k0 + 1];
        bv[0] = W1[(k0 + 0) * 16 + m];               // B 4x16: lane holds W1[k,n=m]
        bv[1] = W1[(k0 + 1) * 16 + m];
        c = __builtin_amdgcn_wmma_f32_16x16x4_f32(
                false, av, false, bv, (short)0, c, false, false);
    }
#pragma unroll
    for (int r = 0; r < 8; ++r) {            // D: VGPR r -> node row r (+8 for hi half)
        const int node = t * 16 + r + 8 * hi;
        const float v  = c[r] * norm_src[node];
        h1[node * 16 + m]   = v;             // gather source
        agg1[node * 16 + m] = v;             // pre-seeded with self-loop term
    }
}

// ------------------------------------------------- edge scatter, 16 features
__global__ void gcn_scatter1(const int* __restrict__ src32,
                             const int* __restrict__ dst32,
                             const float* __restrict__ h1,
                             float* __restrict__ agg1) {
    int tid = blockIdx.x * blockDim.x + threadIdx.x;
    if (tid < kEdges * 16) {
        const int e = tid >> 4;
        const int f = tid & 15;
        const int s = __builtin_nontemporal_load(&src32[e]);
        const int d = __builtin_nontemporal_load(&dst32[e]);
        atomicAdd(&agg1[d * 16 + f], h1[s * 16 + f]);
    }
}

// ------------- layer 2: x = relu(agg1*norm_dst + b1);  (x @ W2pad) * norm_src
__global__ void gcn_layer2_wmma(const float* __restrict__ agg1,
                                const float* __restrict__ bias1,
                                const float* __restrict__ W2,
                                const float* __restrict__ norm_dst,
                                const float* __restrict__ norm_src,
                                float* __restrict__ h2,
                                float* __restrict__ agg2) {
    __shared__ float xs[8][16 * 17];         // pitch 17 -> conflict-free column reads
    const int wave = threadIdx.x >> 5;
    const int lane = threadIdx.x & 31;
    const int t    = blockIdx.x * 8 + wave;
    if (t >= kTiles) return;                 // no cross-wave LDS deps -> no barrier

    // finalize layer-1 output for this tile into LDS (256 elems, 8 per lane)
#pragma unroll
    for (int i = 0; i < 8; ++i) {
        const int idx = lane + 32 * i;
        const int row = idx >> 4;
        const int col = idx & 15;
        float v = agg1[t * 256 + idx] * norm_dst[t * 16 + row] + bias1[col];
        xs[wave][row * 17 + col] = fmaxf(v, 0.0f);   // ReLU
    }

    const int hi = lane >> 4;
    const int m  = lane & 15;
    v8f c = {};
#pragma unroll
    for (int kk = 0; kk < 4; ++kk) {
        const int k0 = kk * 4 + hi * 2;
        v2f av, bv;
        av[0] = xs[wave][m * 17 + k0 + 0];
        av[1] = xs[wave][m * 17 + k0 + 1];
        bv[0] = (m < 8) ? W2[(k0 + 0) * 8 + m] : 0.0f;  // W2 zero-padded to 16 cols
        bv[1] = (m < 8) ? W2[(k0 + 1) * 8 + m] : 0.0f;
        c = __builtin_amdgcn_wmma_f32_16x16x4_f32(
                false, av, false, bv, (short)0, c, false, false);
    }
#pragma unroll
    for (int r = 0; r < 8; ++r) {
        const int node = t * 16 + r + 8 * hi;
        const float v  = c[r] * norm_src[node];
        if (m < 8) {                          // only first 8 output features valid
            h2[node * 8 + m]   = v;
            agg2[node * 8 + m] = v;           // self-loop seed
        }
    }
}

// ------------------------------------------------- edge scatter, 8 features
__global__ void gcn_scatter2(const int* __restrict__ src32,
                             const int* __restrict__ dst32,
                             const float* __restrict__ h2,
                             float* __restrict__ agg2) {
    int tid = blockIdx.x * blockDim.x + threadIdx.x;
    if (tid < kEdges * 8) {
        const int e = tid >> 3;
        const int f = tid & 7;
        const int s = __builtin_nontemporal_load(&src32[e]);
        const int d = __builtin_nontemporal_load(&dst32[e]);
        atomicAdd(&agg2[d * 8 + f], h2[s * 8 + f]);
    }
}

__global__ void gcn_final(const float* __restrict__ agg2,
                          const float* __restrict__ norm_dst,
                          const float* __restrict__ bias2,
                          float* __restrict__ out) {
    int i = blockIdx.x * blockDim.x + threadIdx.x;
    if (i < kNodes * 8) {
        out[i] = agg2[i] * norm_dst[i >> 3] + bias2[i & 7];
    }
}

// ---------------------------------------------------------------- launcher
extern "C" void kernel_launch(void* const* d_in, const int* in_sizes, int n_in,
                              void* d_out, int out_size, void* d_ws, size_t ws_size,
                              hipStream_t stream) {
    const float*     X    = (const float*)d_in[0];      // [100000,16]
    const float*     W1   = (const float*)d_in[1];      // [16,16]
    const float*     b1   = (const float*)d_in[2];      // [16]
    const float*     W2   = (const float*)d_in[3];      // [16,8]
    const float*     b2   = (const float*)d_in[4];      // [8]
    const long long* eidx = (const long long*)d_in[5];  // [2,3200000] int64
    float*           out  = (float*)d_out;              // [100000,8]

    char* p = (char*)d_ws;
    float* deg_out = (float*)p; p += (size_t)kNodes * 4;   // becomes norm_src
    float* deg_in  = (float*)p; p += (size_t)kNodes * 4;   // becomes norm_dst
    int*   src32   = (int*)p;   p += (size_t)kEdges * 4;
    int*   dst32   = (int*)p;   p += (size_t)kEdges * 4;
    float* h1      = (float*)p; p += (size_t)kNodes * 16 * 4;
    float* agg1    = (float*)p; p += (size_t)kNodes * 16 * 4;
    float* h2      = (float*)p; p += (size_t)kNodes * 8 * 4;
    float* agg2    = (float*)p; p += (size_t)kNodes * 8 * 4;

    const int B = 256;
    gcn_init_deg<<<(kNodes + B - 1) / B, B, 0, stream>>>(deg_out, deg_in);
    gcn_prep_edges<<<(kEdges + B - 1) / B, B, 0, stream>>>(eidx, src32, dst32,
                                                           deg_out, deg_in);
    gcn_norm<<<(kNodes + B - 1) / B, B, 0, stream>>>(deg_out, deg_in);

    gcn_layer1_wmma<<<(kTiles + 7) / 8, B, 0, stream>>>(X, W1, deg_out, h1, agg1);
    gcn_scatter1<<<(kEdges * 16 + B - 1) / B, B, 0, stream>>>(src32, dst32, h1, agg1);

    gcn_layer2_wmma<<<(kTiles + 7) / 8, B, 0, stream>>>(agg1, b1, W2, deg_in,
                                                        deg_out, h2, agg2);
    gcn_scatter2<<<(kEdges * 8 + B - 1) / B, B, 0, stream>>>(src32, dst32, h2, agg2);

    gcn_final<<<(kNodes * 8 + B - 1) / B, B, 0, stream>>>(agg2, deg_in, b2, out);
}